// _AnchorTargetLayer_57200374448648
// MI455X (gfx1250) — compile-verified
//
#include <hip/hip_runtime.h>
#include <cstdint>

#define NEG_THR_C 0.3f
#define POS_THR_C 0.7f
#define NUM_FG_C 128
#define ANCHOR_BATCH_C 256

// ---- workspace byte offsets ----
#define WS_GTMAX   0u        // B*K uint (float bits, init 0)
#define WS_CFG     4096u     // B int
#define WS_CBG     4608u     // B int
#define WS_REM     8192u     // 2*B int
#define WS_ACT     8704u     // 2*B int
#define WS_PREFIX  12288u    // 2*B u64
#define WS_THR     13312u    // 2*B u64
#define WS_HIST    16384u    // 2*B*256 u32 (16KB)
#define WS_WVAL    36864u    // float
#define WS_LABELS  65536u    // B*N int8

// ---------- CDNA5 async global->LDS helpers ----------
__device__ __forceinline__ uint32_t lds_addr_of(const void* p) {
  // low 32 bits of a generic pointer to __shared__ == LDS byte offset
  return (uint32_t)(uintptr_t)p;
}
__device__ __forceinline__ void async_copy_f32_g2l(uint32_t lds, const float* g) {
  asm volatile("global_load_async_to_lds_b32 %0, %1, off"
               :: "v"(lds), "v"((uint64_t)(uintptr_t)g) : "memory");
}
__device__ __forceinline__ void wait_async0() {
  asm volatile("s_wait_asynccnt 0" ::: "memory");
}

// ---------- math helpers (no FMA contraction: bitwise reproducible) ----------
__device__ __forceinline__ float iou_one(float ax1, float ay1, float ax2, float ay2,
                                         float a_area,
                                         float gx1, float gy1, float gx2, float gy2,
                                         bool mask_off) {
#pragma clang fp contract(off)
  float gw = gx2 - gx1 + 1.0f, gh = gy2 - gy1 + 1.0f;
  float garea = gw * gh;
  float ix1 = fmaxf(ax1, gx1), iy1 = fmaxf(ay1, gy1);
  float ix2 = fminf(ax2, gx2), iy2 = fminf(ay2, gy2);
  float iw = fmaxf(ix2 - ix1 + 1.0f, 0.0f);
  float ih = fmaxf(iy2 - iy1 + 1.0f, 0.0f);
  float inter = iw * ih;
  float iou = inter / (a_area + garea - inter);
  bool gz = (gw == 1.0f) && (gh == 1.0f);
  return (mask_off || gz) ? 0.0f : iou;
}

__device__ __forceinline__ void anchor_coords(const float* s_an, int a, int wq, int hq,
                                              float& ax1, float& ay1, float& ax2, float& ay2,
                                              float& a_area) {
#pragma clang fp contract(off)
  float shx = (float)wq * 16.0f, shy = (float)hq * 16.0f;
  ax1 = s_an[a * 4 + 0] + shx;
  ay1 = s_an[a * 4 + 1] + shy;
  ax2 = s_an[a * 4 + 2] + shx;
  ay2 = s_an[a * 4 + 3] + shy;
  float aw = ax2 - ax1 + 1.0f, ah = ay2 - ay1 + 1.0f;
  a_area = aw * ah;
}

// ---------- kernels ----------
__global__ void k_init(uint8_t* ws, int words) {
  uint32_t* p = (uint32_t*)ws;
  for (int i = blockIdx.x * blockDim.x + threadIdx.x; i < words; i += gridDim.x * blockDim.x)
    p[i] = 0u;
}

__global__ void k_gtmax(const float* __restrict__ gt, const float* __restrict__ im_info,
                        const float* __restrict__ anch, uint8_t* ws,
                        int B, int K, int A, int W, int H) {
#pragma clang fp contract(off)
  int b = blockIdx.y;
  int N = H * W * A;
  __shared__ float s_gt[128];
  __shared__ float s_an[64];
  __shared__ uint32_t s_max[32];
  int tid = threadIdx.x;
  if (tid < K * 5) async_copy_f32_g2l(lds_addr_of(&s_gt[tid]), gt + (size_t)b * K * 5 + tid);
  if (tid < A * 4) s_an[tid] = anch[tid];
  if (tid < K) s_max[tid] = 0u;
  wait_async0();
  __syncthreads();
  float im_h = im_info[0], im_w = im_info[1];
  int n = blockIdx.x * blockDim.x + tid;
  if (n < N) {
    int a = n % A; int s = n / A; int wq = s % W; int hq = s / W;
    float ax1, ay1, ax2, ay2, aarea;
    anchor_coords(s_an, a, wq, hq, ax1, ay1, ax2, ay2, aarea);
    bool inside = (ax1 >= 0.f) && (ay1 >= 0.f) && (ax2 < im_w) && (ay2 < im_h);
    for (int k = 0; k < K; k++) {
      float iou = iou_one(ax1, ay1, ax2, ay2, aarea,
                          s_gt[k * 5 + 0], s_gt[k * 5 + 1], s_gt[k * 5 + 2], s_gt[k * 5 + 3],
                          !inside);
      atomicMax(&s_max[k], __float_as_uint(iou));
    }
  }
  __syncthreads();
  uint32_t* gm = (uint32_t*)(ws + WS_GTMAX);
  if (tid < K) atomicMax(&gm[b * K + tid], s_max[tid]);
}

__global__ void k_label(const float* __restrict__ gt, const float* __restrict__ im_info,
                        const float* __restrict__ anch, uint8_t* ws,
                        int B, int K, int A, int W, int H) {
#pragma clang fp contract(off)
  int b = blockIdx.y;
  int N = H * W * A;
  __shared__ float s_gt[128];
  __shared__ float s_an[64];
  __shared__ float s_adj[32];
  __shared__ int s_cf, s_cb;
  int tid = threadIdx.x;
  if (tid < K * 5) async_copy_f32_g2l(lds_addr_of(&s_gt[tid]), gt + (size_t)b * K * 5 + tid);
  if (tid < A * 4) s_an[tid] = anch[tid];
  if (tid < K) {
    float g = __uint_as_float(((uint32_t*)(ws + WS_GTMAX))[b * K + tid]);
    s_adj[tid] = (g == 0.0f) ? 1e-5f : g;
  }
  if (tid == 0) { s_cf = 0; s_cb = 0; }
  wait_async0();
  __syncthreads();
  float im_h = im_info[0], im_w = im_info[1];
  int n = blockIdx.x * blockDim.x + tid;
  if (n < N) {
    int a = n % A; int s = n / A; int wq = s % W; int hq = s / W;
    float ax1, ay1, ax2, ay2, aarea;
    anchor_coords(s_an, a, wq, hq, ax1, ay1, ax2, ay2, aarea);
    bool inside = (ax1 >= 0.f) && (ay1 >= 0.f) && (ax2 < im_w) && (ay2 < im_h);
    float max_ov = 0.0f;
    bool keep = false;
    for (int k = 0; k < K; k++) {
      float iou = iou_one(ax1, ay1, ax2, ay2, aarea,
                          s_gt[k * 5 + 0], s_gt[k * 5 + 1], s_gt[k * 5 + 2], s_gt[k * 5 + 3],
                          !inside);
      max_ov = fmaxf(max_ov, iou);
      if (iou == s_adj[k]) keep = true;
    }
    int8_t lab = -1;
    if (inside) {
      if (max_ov < NEG_THR_C) lab = 0;
      if (keep || (max_ov >= POS_THR_C)) lab = 1;
    }
    ((int8_t*)(ws + WS_LABELS))[(size_t)b * N + n] = lab;
    if (lab == 1) atomicAdd(&s_cf, 1);
    else if (lab == 0) atomicAdd(&s_cb, 1);
  }
  __syncthreads();
  if (tid == 0) {
    if (s_cf) atomicAdd(((int*)(ws + WS_CFG)) + b, s_cf);
    if (s_cb) atomicAdd(((int*)(ws + WS_CBG)) + b, s_cb);
  }
}

__global__ void k_prep(uint8_t* ws, int B) {
  int t = threadIdx.x;
  if (t >= 2 * B) return;
  int sel = t / B, b = t % B;
  int cf = ((int*)(ws + WS_CFG))[b];
  int cb = ((int*)(ws + WS_CBG))[b];
  int k, cand;
  if (sel == 0) { k = NUM_FG_C; cand = cf; }
  else {
    int nf = cf < NUM_FG_C ? cf : NUM_FG_C;
    k = ANCHOR_BATCH_C - nf;
    cand = cb;
  }
  int idx = sel * B + b;
  ((uint64_t*)(ws + WS_PREFIX))[idx] = 0ull;
  ((uint64_t*)(ws + WS_THR))[idx] = 0ull;
  if (cand <= k || k <= 0) { ((int*)(ws + WS_ACT))[idx] = 0; ((int*)(ws + WS_REM))[idx] = 0; }
  else { ((int*)(ws + WS_ACT))[idx] = 1; ((int*)(ws + WS_REM))[idx] = k; }
}

__global__ void k_hist(const float* __restrict__ rs, uint8_t* ws, int B, int N, int bytepos) {
  int b = blockIdx.y, sel = blockIdx.z;
  int idx = sel * B + b;
  if (!((int*)(ws + WS_ACT))[idx]) return;   // uniform per block
  __shared__ uint32_t h[256];
  int tid = threadIdx.x;
  h[tid] = 0u;
  __syncthreads();
  uint64_t pref = ((uint64_t*)(ws + WS_PREFIX))[idx];
  uint64_t mask = (bytepos == 7) ? 0ull : (~0ull << (8 * (bytepos + 1)));
  int n = blockIdx.x * blockDim.x + tid;
  if (n < N) {
    int8_t lab = ((int8_t*)(ws + WS_LABELS))[(size_t)b * N + n];
    bool cand = (sel == 0) ? (lab == 1) : (lab == 0);
    if (cand) {
      float sc = rs[((size_t)b * 2 + sel) * N + n];
      uint64_t key = ((uint64_t)__float_as_uint(sc) << 32) | (uint32_t)(~(uint32_t)n);
      if ((key & mask) == pref) {
        int byte = (int)((key >> (8 * bytepos)) & 0xFFull);
        atomicAdd(&h[byte], 1u);
      }
    }
  }
  __syncthreads();
  if (h[tid]) atomicAdd(&((uint32_t*)(ws + WS_HIST))[(size_t)idx * 256 + tid], h[tid]);
}

__global__ void k_pick(uint8_t* ws, int B, int bytepos) {
  int t = threadIdx.x;
  if (t >= 2 * B) return;
  int idx = t;
  int* act = (int*)(ws + WS_ACT);
  if (!act[idx]) return;
  uint32_t* h = &((uint32_t*)(ws + WS_HIST))[(size_t)idx * 256];
  int rem = ((int*)(ws + WS_REM))[idx];
  int cum = 0, v = 255;
  for (; v >= 0; v--) { int c = (int)h[v]; if (cum + c >= rem) break; cum += c; }
  if (v < 0) {
    act[idx] = 0;
    ((uint64_t*)(ws + WS_THR))[idx] = 0ull;  // keep all (should not happen)
  } else {
    ((int*)(ws + WS_REM))[idx] = rem - cum;
    uint64_t p = ((uint64_t*)(ws + WS_PREFIX))[idx] | ((uint64_t)(uint32_t)v << (8 * bytepos));
    ((uint64_t*)(ws + WS_PREFIX))[idx] = p;
    if (bytepos == 0) ((uint64_t*)(ws + WS_THR))[idx] = p;
  }
  for (int i = 0; i < 256; i++) h[i] = 0u;
}

__global__ void k_wval(uint8_t* ws, int B) {
  if (threadIdx.x != 0) return;
  int cf = ((int*)(ws + WS_CFG))[B - 1];
  int cb = ((int*)(ws + WS_CBG))[B - 1];
  int nf = cf < NUM_FG_C ? cf : NUM_FG_C;
  int nbt = ANCHOR_BATCH_C - nf;
  int nb = cb < nbt ? cb : nbt;
  *(float*)(ws + WS_WVAL) = 1.0f / (float)(nf + nb);
}

__global__ void k_out(const float* __restrict__ gt, const float* __restrict__ im_info,
                      const float* __restrict__ anch, const float* __restrict__ rs,
                      const uint8_t* __restrict__ ws, float* __restrict__ out,
                      int B, int K, int A, int W, int H) {
#pragma clang fp contract(off)
  int b = blockIdx.y;
  int tid = threadIdx.x;
  int HWp = H * W;
  int N = HWp * A;
  __shared__ float s_gt[128];
  __shared__ float s_an[64];
  if (tid < K * 5) async_copy_f32_g2l(lds_addr_of(&s_gt[tid]), gt + (size_t)b * K * 5 + tid);
  if (tid < A * 4) s_an[tid] = anch[tid];
  wait_async0();
  __syncthreads();
  float im_h = im_info[0], im_w = im_info[1];
  float wval = *(const float*)(ws + WS_WVAL);
  uint64_t thr_fg = ((const uint64_t*)(ws + WS_THR))[0 * B + b];
  uint64_t thr_bg = ((const uint64_t*)(ws + WS_THR))[1 * B + b];
  int t = blockIdx.x * blockDim.x + tid;  // index in (a, h, w) output order
  if (t >= A * HWp) return;
  int wq = t % W; int r = t / W; int hq = r % H; int a = r / H;
  int n = (hq * W + wq) * A + a;
  float ax1, ay1, ax2, ay2, aarea;
  anchor_coords(s_an, a, wq, hq, ax1, ay1, ax2, ay2, aarea);
  bool inside = (ax1 >= 0.f) && (ay1 >= 0.f) && (ax2 < im_w) && (ay2 < im_h);
  float best = -1.0f; int barg = 0;
  for (int k = 0; k < K; k++) {
    float iou = iou_one(ax1, ay1, ax2, ay2, aarea,
                        s_gt[k * 5 + 0], s_gt[k * 5 + 1], s_gt[k * 5 + 2], s_gt[k * 5 + 3],
                        !inside);
    if (iou > best) { best = iou; barg = k; }   // first occurrence of max kept
  }
  int8_t lab = ((const int8_t*)(ws + WS_LABELS))[(size_t)b * N + n];
  if (lab == 1) {
    float sc = rs[((size_t)b * 2 + 0) * N + n];
    uint64_t key = ((uint64_t)__float_as_uint(sc) << 32) | (uint32_t)(~(uint32_t)n);
    if (key < thr_fg) lab = -1;
  } else if (lab == 0) {
    float sc = rs[((size_t)b * 2 + 1) * N + n];
    uint64_t key = ((uint64_t)__float_as_uint(sc) << 32) | (uint32_t)(~(uint32_t)n);
    if (key < thr_bg) lab = -1;
  }
  // bbox transform vs assigned gt
  float gx1 = s_gt[barg * 5 + 0], gy1 = s_gt[barg * 5 + 1];
  float gx2 = s_gt[barg * 5 + 2], gy2 = s_gt[barg * 5 + 3];
  float ew = ax2 - ax1 + 1.0f, eh = ay2 - ay1 + 1.0f;
  float ecx = ax1 + 0.5f * ew, ecy = ay1 + 0.5f * eh;
  float gw_ = gx2 - gx1 + 1.0f, gh_ = gy2 - gy1 + 1.0f;
  float gcx = gx1 + 0.5f * gw_, gcy = gy1 + 0.5f * gh_;
  float dx = (gcx - ecx) / ew, dy = (gcy - ecy) / eh;
  float dwv = logf(gw_ / ew), dhv = logf(gh_ / eh);
  if (!inside) { dx = 0.f; dy = 0.f; dwv = 0.f; dhv = 0.f; }
  float inw = (lab == 1) ? 1.0f : 0.0f;
  float outw = (lab == 1 || lab == 0) ? wval : 0.0f;
  // write outputs (coalesced per plane)
  size_t HWs = (size_t)HWp;
  size_t p = (size_t)hq * W + wq;
  size_t nl = (size_t)B * A * HWs;
  size_t nt = (size_t)B * 4 * A * HWs;
  float* oL = out;
  float* oT = out + nl;
  float* oI = oT + nt;
  float* oO = oI + nt;
  oL[((size_t)b * A + a) * HWs + p] = (float)lab;
  size_t tb = (((size_t)b * 4 * A) + (size_t)a * 4) * HWs + p;
  oT[tb] = dx; oT[tb + HWs] = dy; oT[tb + 2 * HWs] = dwv; oT[tb + 3 * HWs] = dhv;
  oI[tb] = inw; oI[tb + HWs] = inw; oI[tb + 2 * HWs] = inw; oI[tb + 3 * HWs] = inw;
  oO[tb] = outw; oO[tb + HWs] = outw; oO[tb + 2 * HWs] = outw; oO[tb + 3 * HWs] = outw;
}

extern "C" void kernel_launch(void* const* d_in, const int* in_sizes, int n_in,
                              void* d_out, int out_size, void* d_ws, size_t ws_size,
                              hipStream_t stream) {
  const float* gt = (const float*)d_in[1];
  const float* im = (const float*)d_in[2];
  const float* an = (const float*)d_in[3];
  const float* rs = (const float*)d_in[4];
  float* out = (float*)d_out;
  uint8_t* ws = (uint8_t*)d_ws;

  int B = in_sizes[2] / 3;          // im_info (B,3)
  int A = in_sizes[3] / 4;          // anchors (A,4)
  int K = in_sizes[1] / (5 * B);    // gt_boxes (B,K,5)
  int N = in_sizes[4] / (2 * B);    // rand_scores (B,2,N)
  int HW = N / A;
  int W = 1;
  while ((W + 1) * (W + 1) <= HW) W++;  // isqrt (H==W in this workload)
  int H = HW / W;

  dim3 blk(256);
  dim3 gN((unsigned)((N + 255) / 256), (unsigned)B);

  int zero_words = (int)((WS_WVAL + 256) / 4);
  k_init<<<dim3(40), blk, 0, stream>>>(ws, zero_words);
  k_gtmax<<<gN, blk, 0, stream>>>(gt, im, an, ws, B, K, A, W, H);
  k_label<<<gN, blk, 0, stream>>>(gt, im, an, ws, B, K, A, W, H);
  k_prep<<<dim3(1), dim3(64), 0, stream>>>(ws, B);
  for (int byte = 7; byte >= 0; byte--) {
    k_hist<<<dim3((unsigned)((N + 255) / 256), (unsigned)B, 2), blk, 0, stream>>>(rs, ws, B, N, byte);
    k_pick<<<dim3(1), dim3(64), 0, stream>>>(ws, B, byte);
  }
  k_wval<<<dim3(1), dim3(32), 0, stream>>>(ws, B);
  k_out<<<gN, blk, 0, stream>>>(gt, im, an, rs, ws, out, B, K, A, W, H);
}